// clDiceLoss_21268678050432
// MI455X (gfx1250) — compile-verified
//
#include <hip/hip_runtime.h>
#include <hip/hip_bf16.h>

// ---------------------------------------------------------------------------
// clDice loss, fully fused for MI455X (gfx1250, wave32).
//  - K1: per-tile fused 10-iteration soft-skeleton in LDS (async global->LDS)
//  - K2: WMMA-based reduction of per-tile partial sums
//  - K3: scalar loss
// No atomics anywhere -> deterministic, no workspace zeroing required.
// ---------------------------------------------------------------------------

#define TILE     64
#define HALO     12                 // exact dependency radius is 11
#define TW       (TILE + 2*HALO)    // 88
#define NTHREADS 256
#define PPT      16                 // (64*64)/256 owned pixels per thread
#define NITER    10

typedef __attribute__((ext_vector_type(2))) float v2f;
typedef __attribute__((ext_vector_type(8))) float v8f;

// Low 32 bits of a generic (flat) LDS pointer are the LDS byte offset
// (ISA 10.2: LDS aperture lives entirely in addr[63:32]).
__device__ __forceinline__ unsigned lds_off(const void* p) {
  return (unsigned)(unsigned long long)p;
}

// CDNA5 async global->LDS copy, tracked by ASYNCcnt (CDNA5 ISA).
__device__ __forceinline__ void async_ld_f32(unsigned loff, const float* gp) {
  asm volatile("global_load_async_to_lds_b32 %0, %1, off"
               :: "v"(loff), "v"((unsigned long long)gp)
               : "memory");
}

__device__ __forceinline__ void wait_async0() {
  asm volatile("s_wait_asynccnt 0" ::: "memory");
}

__device__ __forceinline__ float sigmoidf(float x) {
  return 1.0f / (1.0f + __expf(-x));
}

// ---------------------------------------------------------------------------
// K1: fused soft_skel + partial sums per 64x64 tile.
// grid.x = B(16) * C(2) * 16 * 16 tiles; block = 256 threads (8 waves).
// partial layout: partial[(s*32 + (c*16+b))*256 + tileIdx], s in 0..3
//   s0 = sum(skel_pred * target), s1 = sum(skel_pred)
//   s2 = sum(skel_gt   * pred)  , s3 = sum(skel_gt)
// ---------------------------------------------------------------------------
__global__ __launch_bounds__(NTHREADS)
void cldice_tile_kernel(const float* __restrict__ logits,
                        const float* __restrict__ targets,
                        float* __restrict__ partial)
{
  __shared__ float bufA[TW * TW];
  __shared__ float bufB[TW * TW];
  __shared__ float red[8][4];

  const int tid = threadIdx.x;
  const int bid = blockIdx.x;
  const int tX  = bid & 15;
  const int tY  = (bid >> 4) & 15;
  const int ch  = (bid >> 8) & 1;
  const int bb  = bid >> 9;

  const int gx0 = tX * TILE - HALO;
  const int gy0 = tY * TILE - HALO;
  const size_t imgBase = ((size_t)(bb * 2 + ch)) * (size_t)(1024 * 1024);
  const float* logp = logits + imgBase;
  const float* tgtp = targets + imgBase;

  const int tx = tid & 63;   // owned x = HALO + tx
  const int ty = tid >> 6;   // owned y = HALO + 4*j + ty

  float s1 = 0.f, s2 = 0.f, s3 = 0.f, s4 = 0.f;

  for (int pass = 0; pass < 2; ++pass) {
    const float* src = (pass == 0) ? logp : tgtp;

    // ---- async clamped halo-tile load: global -> LDS (ASYNCcnt path) ----
    for (int idx = tid; idx < TW * TW; idx += NTHREADS) {
      int y  = idx / TW;
      int x  = idx - y * TW;
      int gy = gy0 + y; gy = gy < 0 ? 0 : (gy > 1023 ? 1023 : gy);
      int gx = gx0 + x; gx = gx < 0 ? 0 : (gx > 1023 ? 1023 : gx);
      async_ld_f32(lds_off(&bufA[idx]), src + (((size_t)gy) << 10) + gx);
    }
    wait_async0();
    __syncthreads();

    if (pass == 0) {                    // pred = sigmoid(logits), in-place
      for (int idx = tid; idx < TW * TW; idx += NTHREADS)
        bufA[idx] = sigmoidf(bufA[idx]);
      __syncthreads();
    }

    float skel[PPT];
#pragma unroll
    for (int j = 0; j < PPT; ++j) skel[j] = 0.f;

    float* img = bufA;
    float* er  = bufB;

    for (int it = 0; it < NITER; ++it) {
      // erode = min( 3x1 minpool, 1x3 minpool ) over [1,87)^2
      for (int idx = tid; idx < (TW - 2) * (TW - 2); idx += NTHREADS) {
        int y = idx / (TW - 2) + 1;
        int x = idx - (y - 1) * (TW - 2) + 1;
        float c0 = img[y * TW + x];
        float vm = fminf(fminf(img[(y - 1) * TW + x], c0), img[(y + 1) * TW + x]);
        float hm = fminf(fminf(img[y * TW + x - 1],  c0), img[y * TW + x + 1]);
        er[y * TW + x] = fminf(vm, hm);
      }
      __syncthreads();

      // opened = 3x3 maxpool(er) on owned interior pixels; skel += relu(img-opened)
#pragma unroll
      for (int j = 0; j < PPT; ++j) {
        int y = HALO + 4 * j + ty;
        int x = HALO + tx;
        const float* e0 = &er[(y - 1) * TW + x];
        float d = fmaxf(fmaxf(e0[-1], e0[0]), e0[1]);
        const float* e1 = e0 + TW;
        d = fmaxf(d, fmaxf(fmaxf(e1[-1], e1[0]), e1[1]));
        const float* e2 = e1 + TW;
        d = fmaxf(d, fmaxf(fmaxf(e2[-1], e2[0]), e2[1]));
        skel[j] += fmaxf(img[y * TW + x] - d, 0.f);
      }
      __syncthreads();                  // img buffer is overwritten next iter

      float* t = img; img = er; er = t; // img_{i+1} = erode(img_i)
    }

    // multiply skeleton by the "other" operand (interior only, coalesced)
    const float* other = (pass == 0) ? tgtp : logp;
#pragma unroll
    for (int j = 0; j < PPT; ++j) {
      int gy = tY * TILE + 4 * j + ty;
      int gx = tX * TILE + tx;
      float o = other[(((size_t)gy) << 10) + gx];
      if (pass == 1) o = sigmoidf(o);   // tsens uses pred = sigmoid(logits)
      if (pass == 0) { s1 += skel[j] * o; s2 += skel[j]; }
      else           { s3 += skel[j] * o; s4 += skel[j]; }
    }
    __syncthreads();                    // before next pass reuses bufA
  }

  // ---- block reduction: wave32 shuffles + tiny LDS ----
#pragma unroll
  for (int m = 16; m >= 1; m >>= 1) {
    s1 += __shfl_xor(s1, m);
    s2 += __shfl_xor(s2, m);
    s3 += __shfl_xor(s3, m);
    s4 += __shfl_xor(s4, m);
  }
  int wid = tid >> 5;
  if ((tid & 31) == 0) { red[wid][0] = s1; red[wid][1] = s2; red[wid][2] = s3; red[wid][3] = s4; }
  __syncthreads();
  if (tid < 4) {
    float acc = 0.f;
#pragma unroll
    for (int w = 0; w < 8; ++w) acc += red[w][tid];
    int imgIdx  = ch * 16 + bb;
    int tileIdx = tY * 16 + tX;
    partial[(tid * 32 + imgIdx) * 256 + tileIdx] = acc;
  }
}

// ---------------------------------------------------------------------------
// K2: reduce 256 tile partials per (image, sum) with chained WMMA.
// D += ones(16x4) x B(4x16): with A all-ones, every D element is a column sum,
// so four chained v_wmma_f32_16x16x4_f32 reduce 256 values to lane sums
// (layout-agnostic because A is uniform). Finish with a 32-lane shuffle; the
// 16x16 D replicates each column sum twice across the wave -> scale by 0.5.
// grid = 32 (b,c images), block = 32 (one wave; EXEC all ones as WMMA needs).
// ---------------------------------------------------------------------------
__global__ __launch_bounds__(32)
void reduce_wmma_kernel(const float* __restrict__ partial,
                        float* __restrict__ sums)
{
  const int img = blockIdx.x;      // 0..31 = c*16 + b
  const int L   = threadIdx.x;     // 0..31

  v2f a; a.x = 1.0f; a.y = 1.0f;   // ones(16x4)

  for (int s = 0; s < 4; ++s) {
    const float* p = partial + (size_t)(s * 32 + img) * 256;
    v8f c = {};
#pragma unroll
    for (int i = 0; i < 4; ++i) {
      v2f b;
      b.x = p[i * 64 + L];
      b.y = p[i * 64 + 32 + L];
      c = __builtin_amdgcn_wmma_f32_16x16x4_f32(
            /*neg_a=*/false, a, /*neg_b=*/false, b,
            /*c_mod=*/(short)0, c, /*reuse_a=*/false, /*reuse_b=*/false);
    }
    float v = c[0];
#pragma unroll
    for (int m = 16; m >= 1; m >>= 1) v += __shfl_xor(v, m);
    if (L == 0) sums[img * 4 + s] = v * 0.5f;
  }
}

// ---------------------------------------------------------------------------
// K3: loss = 1 - 0.5*(mean_b cl_dice[c=0] + mean_b cl_dice[c=1])
//          = 1 - (1/32) * sum over all 32 (b,c) of cl_dice
// ---------------------------------------------------------------------------
__global__ __launch_bounds__(32)
void finalize_kernel(const float* __restrict__ sums, float* __restrict__ out)
{
  const int t = threadIdx.x;       // 0..31
  const float eps = 1e-6f;
  float S1 = sums[t * 4 + 0];
  float S2 = sums[t * 4 + 1];
  float S3 = sums[t * 4 + 2];
  float S4 = sums[t * 4 + 3];
  float tprec = S1 / (S2 + eps);
  float tsens = S3 / (S4 + eps);
  float cl = 2.f * tprec * tsens / (tprec + tsens + eps);
#pragma unroll
  for (int m = 16; m >= 1; m >>= 1) cl += __shfl_xor(cl, m);
  if (t == 0) out[0] = 1.0f - cl * (1.0f / 32.0f);
}

// ---------------------------------------------------------------------------
extern "C" void kernel_launch(void* const* d_in, const int* in_sizes, int n_in,
                              void* d_out, int out_size, void* d_ws, size_t ws_size,
                              hipStream_t stream) {
  const float* logits  = (const float*)d_in[0];
  const float* targets = (const float*)d_in[1];

  float* partial = (float*)d_ws;           // 4 * 32 * 256 floats = 128 KB
  float* sums    = partial + 4 * 32 * 256; // 32 * 4 floats

  // 16 batch * 2 ch * 16x16 tiles = 8192 workgroups, fully overwrite partial[]
  cldice_tile_kernel<<<dim3(16 * 2 * 16 * 16), dim3(NTHREADS), 0, stream>>>(
      logits, targets, partial);
  reduce_wmma_kernel<<<dim3(32), dim3(32), 0, stream>>>(partial, sums);
  finalize_kernel<<<dim3(1), dim3(32), 0, stream>>>(sums, (float*)d_out);
}